// _RFFTDecompWithGMAC_19490561590127
// MI455X (gfx1250) — compile-verified
//
#include <hip/hip_runtime.h>

typedef __attribute__((ext_vector_type(2))) float v2f;
typedef __attribute__((ext_vector_type(8))) float v8f;

#define NFFT   512
#define SPLIT0 257
#define RBATCH 32768

// Each wave (32 lanes) processes whole batches independently:
//   lane (p = lane&15, h = lane>>4)
//   A-matrix (16x4 f32 WMMA frag, chunk c): lane holds y[m=p, l=4c+2h+{0,1}]
//   where y[m,l] = x[b, 16*l + bitrev4(m)]  -> contiguous 64B blocks per half-wave.
__global__ __launch_bounds__(256) void rfft512_wmma_kernel(
    const float* __restrict__ x,
    float* __restrict__ outRe,
    float* __restrict__ outIm,
    int nWaves)
{
  const int lane = threadIdx.x & 31;
  const int wIdx = (int)(blockIdx.x * (blockDim.x >> 5)) + (threadIdx.x >> 5);
  const int p = lane & 15;
  const int h = lane >> 4;
  const int rb = ((p & 1) << 3) | ((p & 2) << 1) | ((p & 4) >> 1) | ((p & 8) >> 3);

  const float TWO_PI = 6.28318530717958647692f;

  // ---- DFT-32 matrix fragments (B operand), per-lane constants ----
  // B frag (4x16, chunk c, n-tile r): component d holds F[l = 4c+2h+d][k = 16r+p]
  v2f Br[8][2], Bi[8][2];
#pragma unroll
  for (int c = 0; c < 8; ++c)
#pragma unroll
    for (int r = 0; r < 2; ++r)
#pragma unroll
      for (int d = 0; d < 2; ++d) {
        int lB = 4 * c + 2 * h + d;
        int kB = 16 * r + p;
        float ang = -TWO_PI * (float)((lB * kB) & 31) * (1.0f / 32.0f);
        float s, cs;
        __sincosf(ang, &s, &cs);
        Br[c][r][d] = cs;
        Bi[c][r][d] = s;
      }

  // ---- Twiddles (functions of lane only). T4[q] = w512^(p+16q); T3=T4^2; T2=T3^2; T1=T2^2
  float T4r[16], T4i[16], T3r[8], T3i[8], T2r[4], T2i[4], T1r[2], T1i[2];
#pragma unroll
  for (int q = 0; q < 16; ++q) {
    float ang = -TWO_PI * (float)(p + 16 * q) * (1.0f / 512.0f);
    __sincosf(ang, &T4i[q], &T4r[q]);
  }
#pragma unroll
  for (int q = 0; q < 8; ++q) { T3r[q] = T4r[q]*T4r[q] - T4i[q]*T4i[q]; T3i[q] = 2.0f*T4r[q]*T4i[q]; }
#pragma unroll
  for (int q = 0; q < 4; ++q) { T2r[q] = T3r[q]*T3r[q] - T3i[q]*T3i[q]; T2i[q] = 2.0f*T3r[q]*T3i[q]; }
#pragma unroll
  for (int q = 0; q < 2; ++q) { T1r[q] = T2r[q]*T2r[q] - T2i[q]*T2i[q]; T1i[q] = 2.0f*T2r[q]*T2i[q]; }

  float* outBase = h ? outIm : outRe;   // low half-wave -> real plane, high -> imag plane

  for (int b = wIdx; b < RBATCH; b += nWaves) {
    const float* rowp = x + (size_t)b * NFFT + 32 * h + rb;

    // ---- DFT-32 via WMMA f32 16x16x4: z[m,k] = sum_l y[m,l] * F[l,k]
    v8f zr0 = {}, zi0 = {}, zr1 = {}, zi1 = {};
#pragma unroll
    for (int c = 0; c < 8; ++c) {
      v2f a;
      a[0] = __builtin_nontemporal_load(rowp + 64 * c);        // l = 4c+2h
      a[1] = __builtin_nontemporal_load(rowp + 64 * c + 16);   // l = 4c+2h+1
      zr0 = __builtin_amdgcn_wmma_f32_16x16x4_f32(false, a, false, Br[c][0], (short)0, zr0, false, false);
      zi0 = __builtin_amdgcn_wmma_f32_16x16x4_f32(false, a, false, Bi[c][0], (short)0, zi0, false, false);
      zr1 = __builtin_amdgcn_wmma_f32_16x16x4_f32(false, a, false, Br[c][1], (short)0, zr1, false, false);
      zi1 = __builtin_amdgcn_wmma_f32_16x16x4_f32(false, a, false, Bi[c][1], (short)0, zi1, false, false);
    }
    // Lane now holds z[m = v+8h, k = p] in zr0/zi0[v] and z[m, k = 16+p] in zr1/zi1[v].

    // ---- Stage 1: combine (m=2j, 2j+1) -> 4 seqs of 64 per half-wave. Position t = p+16q.
    float a1r[4][4], a1i[4][4];
#pragma unroll
    for (int j = 0; j < 4; ++j) {
      float er = zr0[2*j], ei = zi0[2*j], odr = zr0[2*j+1], odi = zi0[2*j+1];
      float tr = T1r[0]*odr - T1i[0]*odi;
      float ti = T1r[0]*odi + T1i[0]*odr;
      a1r[j][0] = er + tr; a1i[j][0] = ei + ti;
      a1r[j][2] = er - tr; a1i[j][2] = ei - ti;
      er = zr1[2*j]; ei = zi1[2*j]; odr = zr1[2*j+1]; odi = zi1[2*j+1];
      tr = T1r[1]*odr - T1i[1]*odi;
      ti = T1r[1]*odi + T1i[1]*odr;
      a1r[j][1] = er + tr; a1i[j][1] = ei + ti;
      a1r[j][3] = er - tr; a1i[j][3] = ei - ti;
    }

    // ---- Stage 2: 2 seqs of 128 per half-wave.
    float a2r[2][8], a2i[2][8];
#pragma unroll
    for (int j = 0; j < 2; ++j)
#pragma unroll
      for (int q = 0; q < 4; ++q) {
        float er = a1r[2*j][q], ei = a1i[2*j][q];
        float odr = a1r[2*j+1][q], odi = a1i[2*j+1][q];
        float tr = T2r[q]*odr - T2i[q]*odi;
        float ti = T2r[q]*odi + T2i[q]*odr;
        a2r[j][q]     = er + tr; a2i[j][q]     = ei + ti;
        a2r[j][q + 4] = er - tr; a2i[j][q + 4] = ei - ti;
      }

    // ---- Stage 3: 1 seq of 256 per half-wave (h=0 -> even seq, h=1 -> odd seq).
    float a3r[16], a3i[16];
#pragma unroll
    for (int q = 0; q < 8; ++q) {
      float er = a2r[0][q], ei = a2i[0][q];
      float odr = a2r[1][q], odi = a2i[1][q];
      float tr = T3r[q]*odr - T3i[q]*odi;
      float ti = T3r[q]*odi + T3i[q]*odr;
      a3r[q]     = er + tr; a3i[q]     = ei + ti;
      a3r[q + 8] = er - tr; a3i[q + 8] = ei - ti;
    }

    // ---- Stage 4: cross half-wave exchange + final butterfly; store bins 0..256.
    float* ob = outBase + (size_t)b * SPLIT0;
#pragma unroll
    for (int q = 0; q < 16; ++q) {
      float xr = __shfl_xor(a3r[q], 16, 32);
      float xi = __shfl_xor(a3i[q], 16, 32);
      float s0r = h ? xr : a3r[q];
      float s0i = h ? xi : a3i[q];
      float s1r = h ? a3r[q] : xr;
      float s1i = h ? a3i[q] : xi;
      float tr = T4r[q]*s1r - T4i[q]*s1i;
      float ti = T4r[q]*s1i + T4i[q]*s1r;
      float Xr = s0r + tr;
      float Xi = s0i + ti;
      __builtin_nontemporal_store(h ? Xi : Xr, ob + p + 16 * q);  // t = p+16q in [0,255]
      if (q == 0 && p == 0) {
        // bin 256: X[256] = seq0[0] - seq1[0]; lane0 -> real, lane16 -> imag
        __builtin_nontemporal_store(h ? (s0i - s1i) : (s0r - s1r), ob + 256);
      }
    }
  }
}

extern "C" void kernel_launch(void* const* d_in, const int* in_sizes, int n_in,
                              void* d_out, int out_size, void* d_ws, size_t ws_size,
                              hipStream_t stream) {
  (void)in_sizes; (void)n_in; (void)d_ws; (void)ws_size; (void)out_size;
  const float* x = (const float*)d_in[0];
  float* outRe = (float*)d_out;
  float* outIm = outRe + (size_t)RBATCH * SPLIT0;

  const int block = 256;                 // 8 wave32 per workgroup
  const int grid  = 512;                 // 4096 waves -> 8 batches per wave
  const int nWaves = grid * (block / 32);
  rfft512_wmma_kernel<<<grid, block, 0, stream>>>(x, outRe, outIm, nWaves);
}